// ClusterAssigner_7859790152237
// MI455X (gfx1250) — compile-verified
//
#include <hip/hip_runtime.h>

// ---------------- problem constants (match reference) ----------------
#define NPTS   8000
#define GXc    401
#define GYc    401
#define GZc    2
#define DG     (GXc*GYc*GZc)        // 321602 dense voxel cells
#define NTILE  (NPTS/16)            // 500 tiles of 16
#define NSTRIP (NTILE/4)            // 125 strips of 64 columns
#define CONN2  0.36f                // 0.6^2
#define ITERS  64

typedef __attribute__((ext_vector_type(2))) float v2f;
typedef __attribute__((ext_vector_type(8))) float v8f;

// ---------------- zero init (present grid + accumulators) ----------------
__global__ void kzero(int* present, float* counts, float* sx, float* sy) {
    int g = blockIdx.x * 256 + threadIdx.x;
    if (g < DG) present[g] = 0;
    if (g < NPTS) { counts[g] = 0.f; sx[g] = 0.f; sy[g] = 0.f; }
}

// ---------------- voxel id per point + presence mark ----------------
__global__ void kvid(const float* __restrict__ pts, int* __restrict__ present,
                     int* __restrict__ vid) {
    int p = blockIdx.x * 256 + threadIdx.x;
    if (p >= NPTS) return;
    float x = pts[3*p], y = pts[3*p+1], z = pts[3*p+2];
    int cx = (int)floorf((x + 50.f) * 4.0f);          // /0.25
    int cy = (int)floorf((y + 50.f) * 4.0f);
    int cz = (int)floorf((z + 3.f) * (1.0f/6.0f));
    cx = min(max(cx, 0), GXc-1);
    cy = min(max(cy, 0), GYc-1);
    cz = min(max(cz, 0), GZc-1);
    int v = (cx * GYc + cy) * GZc + cz;               // batch==0
    vid[p] = v;
    present[v] = 1;
}

// ---------------- generic 2-level prefix scan ----------------
__global__ void scan_block(const int* __restrict__ in, int* __restrict__ out,
                           int* __restrict__ part, int n) {
    __shared__ int sh[256];
    int g = blockIdx.x * 256 + threadIdx.x;
    int v = (g < n) ? in[g] : 0;
    sh[threadIdx.x] = v;
    __syncthreads();
    for (int off = 1; off < 256; off <<= 1) {
        int t = (threadIdx.x >= off) ? sh[threadIdx.x - off] : 0;
        __syncthreads();
        sh[threadIdx.x] += t;
        __syncthreads();
    }
    if (g < n) out[g] = sh[threadIdx.x];
    if (threadIdx.x == 255) part[blockIdx.x] = sh[255];
}
__global__ void scan_partials(int* part, int nb, int* total) {
    if (blockIdx.x == 0 && threadIdx.x == 0) {
        int acc = 0;
        for (int b = 0; b < nb; b++) { int t = part[b]; part[b] = acc; acc += t; }
        if (total) *total = acc;
    }
}
__global__ void scan_add(int* out, const int* __restrict__ part, int n) {
    int g = blockIdx.x * 256 + threadIdx.x;
    if (g < n) out[g] += part[blockIdx.x];
}

// ---------------- inverse index + segment sums ----------------
__global__ void kinv(const float* __restrict__ pts, const int* __restrict__ vid,
                     const int* __restrict__ scan, int* __restrict__ inv,
                     float* counts, float* sx, float* sy) {
    int p = blockIdx.x * 256 + threadIdx.x;
    if (p >= NPTS) return;
    int u = scan[vid[p]] - 1;                 // sorted-unique rank (inclusive scan - 1)
    inv[p] = u;
    atomicAdd(&counts[u], 1.0f);
    atomicAdd(&sx[u], pts[3*p]);
    atomicAdd(&sy[u], pts[3*p+1]);
}

// ---------------- centers, squared norms, label init (both buffers) ----------------
__global__ void kcent(const float* __restrict__ counts, const float* __restrict__ sx,
                      const float* __restrict__ sy, const int* __restrict__ Mptr,
                      float* X, float* Y, float* SQ, int* LA, int* LB) {
    int u = blockIdx.x * 256 + threadIdx.x;
    if (u >= NPTS) return;
    int M = *Mptr;
    float x = 0.f, y = 0.f;
    if (u < M) {
        float c = fmaxf(counts[u], 1.0f);
        x = sx[u] / c; y = sy[u] / c;
    }
    X[u] = x; Y[u] = y; SQ[u] = x*x + y*y;
    LA[u] = u; LB[u] = u;
}

// ---------------- WMMA adjacency: 16x64 strip per wave (4 WMMAs, A reused) ----
// A(16x4 f32): lanes 0-15 hold {K0,K1}={x,y}; lanes 16-31 hold {K2,K3}=0 pad.
// B(4x16 f32): lanes 0-15 hold {K0,K1}; lanes 16-31 hold {K2,K3}=0 pad.
// D(16x16 f32): VGPR r -> row r (lanes 0-15) / row r+8 (lanes 16-31), col = lane&15.
// Each row's 64 adjacency bits for the strip are packed into one aligned b64 store.
__global__ __launch_bounds__(160) void kadj(const float* __restrict__ X,
                                            const float* __restrict__ Y,
                                            const float* __restrict__ SQ,
                                            const int* __restrict__ Mptr,
                                            unsigned long long* __restrict__ adj64) {
    const int lane   = threadIdx.x;                    // 0..31
    const int strip  = blockIdx.x * 5 + threadIdx.y;   // 0..124 (64 columns each)
    const int tile_i = blockIdx.y;                     // 0..499
    const int i0 = tile_i * 16;
    const int M  = *Mptr;

    v2f a = {0.f, 0.f};
    if (lane < 16) { a.x = X[i0 + lane]; a.y = Y[i0 + lane]; }

    float sqi[8];
#pragma unroll
    for (int r = 0; r < 8; r++)
        sqi[r] = SQ[i0 + ((lane < 16) ? r : r + 8)];

    int myrow = -1;
    if (lane < 8)                     myrow = i0 + lane;
    else if (lane >= 16 && lane < 24) myrow = i0 + 8 + (lane - 16);

    unsigned long long acc = 0ull;
#pragma unroll
    for (int t = 0; t < 4; t++) {
        const int j0 = (strip * 4 + t) * 16;
        v2f b = {0.f, 0.f};
        if (lane < 16) { b.x = X[j0 + lane]; b.y = Y[j0 + lane]; }
        v8f c = {};
        c = __builtin_amdgcn_wmma_f32_16x16x4_f32(false, a, false, b,
                                                  (short)0, c, false, false);
        const int gj  = j0 + (lane & 15);
        const bool vj = gj < M;
        const float sqj = SQ[gj];
#pragma unroll
        for (int r = 0; r < 8; r++) {
            int gi = i0 + ((lane < 16) ? r : r + 8);
            bool conn = ((gi < M) && vj && (sqi[r] + sqj - 2.0f * c[r] < CONN2))
                        || (gi == gj);
            unsigned bal = __builtin_amdgcn_ballot_w32(conn);
            if (lane == r)      acc |= (unsigned long long)(bal & 0xffffu) << (16 * t);
            if (lane == 16 + r) acc |= (unsigned long long)(bal >> 16)     << (16 * t);
        }
    }
    if (myrow >= 0) adj64[(size_t)myrow * NSTRIP + strip] = acc;  // row stride 1000 B
}

// ------- neighbor-min over adjacency bitmask + fused pointer jump (1 wave/row) ----
// Ld entries are always valid in-component labels (init u, monotone decreasing),
// so racy jump reads only accelerate convergence to the unique fixpoint.
__global__ __launch_bounds__(256) void krowmin(const unsigned* __restrict__ adj32,
                                               const int* __restrict__ Ls,
                                               int* __restrict__ Ld) {
    const int row  = blockIdx.x * 8 + threadIdx.y;
    const int lane = threadIdx.x;
    const unsigned* w = adj32 + (size_t)row * (NTILE/2);
    int m = 0x7fffffff;
    for (int t = lane; t < NTILE/2; t += 32) {
        unsigned word = w[t];
        int base = t * 32;
        while (word) {
            int bpos = __ffs(word) - 1;
            word &= word - 1;
            m = min(m, Ls[base + bpos]);
        }
    }
    for (int off = 16; off; off >>= 1) m = min(m, __shfl_xor(m, off, 32));
    if (lane == 0) {
        m = min(m, Ld[m]);     // pointer jump (fresh or stale -> still valid label)
        m = min(m, Ld[m]);
        Ld[row] = m;
    }
}

// ---------------- component root flags ----------------
__global__ void kflags(const int* __restrict__ L, const int* __restrict__ Mptr,
                       int* flags) {
    int i = blockIdx.x * 256 + threadIdx.x;
    if (i >= NPTS) return;
    int M = *Mptr;
    flags[i] = (i < M && L[i] == i) ? 1 : 0;
}

// ---------------- final per-point output ----------------
// out (int32 view): [NPTS x 3] cluster_inds_per_point, then [NPTS] valid_mask (=1)
__global__ void kout(const int* __restrict__ inv, const int* __restrict__ L,
                     const int* __restrict__ fscan, int* __restrict__ out) {
    int p = blockIdx.x * 256 + threadIdx.x;
    if (p >= NPTS) return;
    int u = inv[p];
    int comp = fscan[L[u]] - 1;       // rank of root among sorted roots
    out[p*3 + 0] = 0;                 // cls
    out[p*3 + 1] = 0;                 // batch_idx
    out[p*3 + 2] = comp;              // cluster id
    out[3*NPTS + p] = 1;              // valid_mask true
}

// ---------------- workspace layout (4-byte words) ----------------
static constexpr int o_present = 0;
static constexpr int o_scan    = o_present + DG;
static constexpr int o_part    = o_scan + DG;
static constexpr int o_M       = o_part + 2048;
static constexpr int o_vid     = o_M + 8;
static constexpr int o_inv     = o_vid + NPTS;
static constexpr int o_counts  = o_inv + NPTS;
static constexpr int o_sumx    = o_counts + NPTS;
static constexpr int o_sumy    = o_sumx + NPTS;
static constexpr int o_X       = o_sumy + NPTS;
static constexpr int o_Y       = o_X + NPTS;
static constexpr int o_SQ      = o_Y + NPTS;
static constexpr int o_LA      = o_SQ + NPTS;
static constexpr int o_LB      = o_LA + NPTS;
static constexpr int o_flags   = o_LB + NPTS;
static constexpr int o_fscan   = o_flags + NPTS;
static constexpr int o_adj     = o_fscan + NPTS;     // 2,000,000 words follow (8 MB)

extern "C" void kernel_launch(void* const* d_in, const int* in_sizes, int n_in,
                              void* d_out, int out_size, void* d_ws, size_t ws_size,
                              hipStream_t stream) {
    const float* pts = (const float*)d_in[0];
    (void)in_sizes; (void)n_in; (void)out_size; (void)ws_size;

    int*   ws      = (int*)d_ws;
    int*   present = ws + o_present;
    int*   scan    = ws + o_scan;
    int*   part    = ws + o_part;
    int*   Mptr    = ws + o_M;
    int*   vid     = ws + o_vid;
    int*   inv     = ws + o_inv;
    float* counts  = (float*)(ws + o_counts);
    float* sumx    = (float*)(ws + o_sumx);
    float* sumy    = (float*)(ws + o_sumy);
    float* X       = (float*)(ws + o_X);
    float* Y       = (float*)(ws + o_Y);
    float* SQ      = (float*)(ws + o_SQ);
    int*   LA      = ws + o_LA;
    int*   LB      = ws + o_LB;
    int*   flags   = ws + o_flags;
    int*   fscan   = ws + o_fscan;
    unsigned long long* adj64 = (unsigned long long*)(ws + o_adj);
    unsigned*           adj32 = (unsigned*)(ws + o_adj);

    const int NB_DG = (DG + 255) / 256;       // 1257
    const int NB_N  = (NPTS + 255) / 256;     // 32

    kzero<<<NB_DG, 256, 0, stream>>>(present, counts, sumx, sumy);
    kvid<<<NB_N, 256, 0, stream>>>(pts, present, vid);

    // sorted-unique ranks via dense prefix scan
    scan_block<<<NB_DG, 256, 0, stream>>>(present, scan, part, DG);
    scan_partials<<<1, 256, 0, stream>>>(part, NB_DG, Mptr);
    scan_add<<<NB_DG, 256, 0, stream>>>(scan, part, DG);

    kinv<<<NB_N, 256, 0, stream>>>(pts, vid, scan, inv, counts, sumx, sumy);
    kcent<<<NB_N, 256, 0, stream>>>(counts, sumx, sumy, Mptr, X, Y, SQ, LA, LB);

    // WMMA pairwise distances -> bit-packed adjacency (16x64 strip per wave)
    kadj<<<dim3(NSTRIP/5, NTILE), dim3(32, 5), 0, stream>>>(X, Y, SQ, Mptr, adj64);

    // fixed-count label propagation with fused pointer jumping (deterministic fixpoint)
    int* src = LA; int* dst = LB;
    for (int it = 0; it < ITERS; it++) {
        krowmin<<<NPTS/8, dim3(32, 8), 0, stream>>>(adj32, src, dst);
        int* t = src; src = dst; dst = t;
    }
    // ITERS even -> final labels back in LA (== src)

    kflags<<<NB_N, 256, 0, stream>>>(src, Mptr, flags);
    scan_block<<<NB_N, 256, 0, stream>>>(flags, fscan, part, NPTS);
    scan_partials<<<1, 256, 0, stream>>>(part, NB_N, nullptr);
    scan_add<<<NB_N, 256, 0, stream>>>(fscan, part, NPTS);

    kout<<<NB_N, 256, 0, stream>>>(inv, src, fscan, (int*)d_out);
}